// SampleConditionalGMM_76579266887896
// MI455X (gfx1250) — compile-verified
//
#include <hip/hip_runtime.h>
#include <cstdint>
#include <cstddef>

// Native clang vector types (HIP float4 is a struct; builtins want real vectors)
typedef float f32x4 __attribute__((ext_vector_type(4)));
typedef int   i32x4 __attribute__((ext_vector_type(4)));

#define TABLE_LDS_DWORDS 1024  // holds L + 2*L*C dwords (needs 100 for L=20,C=2)

__global__ void gmm_sample_kernel(
    const int*   __restrict__ labels,      // [B, V]
    const float* __restrict__ means,       // [B, L, C]
    const float* __restrict__ stds,        // [B, L, C]
    const float* __restrict__ noise,       // [B, C, V]
    const int*   __restrict__ gen_labels,  // [L] sorted
    float*       __restrict__ out,         // [B, C, V]
    int V, int L, int C)
{
    __shared__ int s_tab[TABLE_LDS_DWORDS];
    const int b   = blockIdx.y;
    const int tid = threadIdx.x;
    const int LC  = L * C;

    // ---- stage tiny per-batch tables into LDS via CDNA5 async-to-LDS path ----
    // layout: s_tab[0..L) = gen_labels, [L..L+LC) = means[b], [L+LC..L+2LC) = stds[b]
    if (tid < L) {
        uint32_t lds = (uint32_t)(uintptr_t)&s_tab[tid];
        uint64_t g   = (uint64_t)(uintptr_t)(gen_labels + tid);
        asm volatile("global_load_async_to_lds_b32 %0, %1, off"
                     :: "v"(lds), "v"(g) : "memory");
    }
    if (tid < LC) {
        uint32_t lds_m = (uint32_t)(uintptr_t)&s_tab[L + tid];
        uint64_t g_m   = (uint64_t)(uintptr_t)(means + (size_t)b * LC + tid);
        asm volatile("global_load_async_to_lds_b32 %0, %1, off"
                     :: "v"(lds_m), "v"(g_m) : "memory");
        uint32_t lds_s = (uint32_t)(uintptr_t)&s_tab[L + LC + tid];
        uint64_t g_s   = (uint64_t)(uintptr_t)(stds + (size_t)b * LC + tid);
        asm volatile("global_load_async_to_lds_b32 %0, %1, off"
                     :: "v"(lds_s), "v"(g_s) : "memory");
    }
    asm volatile("s_wait_asynccnt 0" ::: "memory");
    __syncthreads();

    const int*   s_gl   = s_tab;
    const float* s_mean = (const float*)&s_tab[L];
    const float* s_std  = (const float*)&s_tab[L + LC];

    const long long v0 = ((long long)blockIdx.x * (long long)blockDim.x + tid) * 4;
    if (v0 >= V) return;

    if (v0 + 3 < (long long)V) {
        // ---- fast path: 4 voxels, 128-bit accesses, labels loaded once ----
        const i32x4 lab4 =
            __builtin_nontemporal_load((const i32x4*)(labels + (size_t)b * V + v0));
        int  idx[4];
        bool val[4];
        #pragma unroll
        for (int e = 0; e < 4; ++e) {
            const int lab = lab4[e];
            int lo = 0, hi = L;                 // searchsorted (side='left')
            while (lo < hi) {
                int mid = (lo + hi) >> 1;
                if (s_gl[mid] < lab) lo = mid + 1; else hi = mid;
            }
            int i = lo < (L - 1) ? lo : (L - 1);   // clip to [0, L-1]
            idx[e] = i;
            val[e] = (s_gl[i] == lab);
        }
        for (int c = 0; c < C; ++c) {
            const size_t off = ((size_t)b * C + c) * (size_t)V + v0;
            const f32x4 n4 = __builtin_nontemporal_load((const f32x4*)(noise + off));
            f32x4 o4;
            #pragma unroll
            for (int e = 0; e < 4; ++e) {
                const float m = s_mean[idx[e] * C + c];
                const float s = s_std [idx[e] * C + c];
                o4[e] = val[e] ? fmaf(s, n4[e], m) : 0.0f;
            }
            __builtin_nontemporal_store(o4, (f32x4*)(out + off));
        }
    } else {
        // ---- scalar tail (only the final partial group of <4 voxels) ----
        for (long long v = v0; v < (long long)V; ++v) {
            const int lab = labels[(size_t)b * V + v];
            int lo = 0, hi = L;
            while (lo < hi) {
                int mid = (lo + hi) >> 1;
                if (s_gl[mid] < lab) lo = mid + 1; else hi = mid;
            }
            int i = lo < (L - 1) ? lo : (L - 1);
            const bool val = (s_gl[i] == lab);
            for (int c = 0; c < C; ++c) {
                const size_t off = ((size_t)b * C + c) * (size_t)V + v;
                out[off] = val ? fmaf(s_std[i * C + c], noise[off], s_mean[i * C + c])
                               : 0.0f;
            }
        }
    }
}

extern "C" void kernel_launch(void* const* d_in, const int* in_sizes, int n_in,
                              void* d_out, int out_size, void* d_ws, size_t ws_size,
                              hipStream_t stream) {
    const int*   labels     = (const int*)  d_in[0];  // [B,1,D,H,W] int32
    const float* means      = (const float*)d_in[1];  // [B,L,C]
    const float* stds       = (const float*)d_in[2];  // [B,L,C]
    const float* noise      = (const float*)d_in[3];  // [B,C,D,H,W]
    const int*   gen_labels = (const int*)  d_in[4];  // [L]
    float*       out        = (float*)d_out;          // [B,C,D,H,W]

    const int L     = in_sizes[4];               // 20
    const int nLab  = in_sizes[0];               // B*V
    const int nNoi  = in_sizes[3];               // B*C*V
    const int C     = nNoi / nLab;               // 2
    const int B     = in_sizes[1] / (L * C);     // 2
    const int V     = nLab / B;                  // D*H*W = 4,096,000

    const int threads = 256;                     // 8 wave32 waves
    const int groups  = (V + 3) / 4;             // vec4 groups per batch
    const int blocksX = (groups + threads - 1) / threads;
    dim3 grid(blocksX, B, 1);

    hipLaunchKernelGGL(gmm_sample_kernel, grid, dim3(threads, 1, 1), 0, stream,
                       labels, means, stds, noise, gen_labels, out, V, L, C);
}